// HONU_60189671686531
// MI455X (gfx1250) — compile-verified
//
#include <hip/hip_runtime.h>

typedef __attribute__((ext_vector_type(2))) float v2f;
typedef __attribute__((ext_vector_type(8))) float v8f;

#define NFEAT        129                 // 128 features + bias column
#define KPAD         132                 // K padded to multiple of 4
#define BATCH        256
#define NPAIRS       8385                // C(130,2) pairs (i1<=i2)
#define NPAIRS_PAD   8400                // padded to multiple of 16
#define NTILES       (NPAIRS_PAD / 16)   // 525 N-tiles
#define MTILES       (BATCH / 16)        // 16 M-tiles
#define M_PER_WAVE   2                   // M-tiles per wave (B reuse)
#define MGROUPS      (MTILES / M_PER_WAVE)   // 8
#define NT_PER_WAVE  5                   // N-tiles per wave
#define NGROUPS      (NTILES / NT_PER_WAVE)  // 105

// workspace layout (in floats)
#define XB_OFF    0
#define W_OFF     (BATCH * KPAD)                     // 33,792
#define PT_OFF    (W_OFF + NPAIRS_PAD * KPAD)        // + 1,108,800 (pairTab as int)
#define PART_OFF  (PT_OFF + NPAIRS_PAD)              // + 8,400
#define WS_FLOATS (PART_OFF + NGROUPS * BATCH)       // + 26,880  -> ~4.71 MB

// ---------------------------------------------------------------- zero ws
__global__ void honu_zero(float* __restrict__ p, long n) {
    long i = (long)blockIdx.x * blockDim.x + threadIdx.x;
    if (i < n) p[i] = 0.0f;
}

// ---------------------------------------------------------------- build xb = [1 | x | 0-pad]
__global__ void honu_build_xb(const float* __restrict__ x, float* __restrict__ xb) {
    const int b = blockIdx.x;      // batch row
    const int j = threadIdx.x;     // 0..KPAD-1
    float v = 0.0f;
    if (j == 0)            v = 1.0f;
    else if (j < NFEAT)    v = x[b * (NFEAT - 1) + (j - 1)];
    xb[(size_t)b * KPAD + j] = v;
}

// ---------------------------------------------------------------- scatter weights into W[p][i0]
__global__ void honu_scatter(const int* __restrict__ cidx,
                             const float* __restrict__ w,
                             float* __restrict__ Wm,
                             int* __restrict__ pairTab,
                             int nM) {
    const int m = blockIdx.x * blockDim.x + threadIdx.x;
    if (m >= nM) return;
    const int i0 = cidx[3 * m + 0];
    const int i1 = cidx[3 * m + 1];
    const int i2 = cidx[3 * m + 2];
    // lexicographic pair index over (i1<=i2) in [0, NFEAT)
    const int p = (i1 * (2 * NFEAT - i1 + 1)) / 2 + (i2 - i1);
    Wm[(size_t)p * KPAD + i0] = w[m];
    pairTab[p] = i1 | (i2 << 16);
}

// ---------------------------------------------------------------- WMMA GEMM + fused pair epilogue
// One wave per block, occupancy-1 (full register file) so that both
// M-tiles' A fragments stay register-resident while B streams from L2.
// Computes NT_PER_WAVE 16-wide column strips of
// S = xb (256xKPAD) * W^T (KPADxNPAIRS_PAD) for TWO 16-row M-tiles that
// share every B fragment (16 FLOP per L2 byte). Each S element is scaled
// by xb[b,i1(p)]*xb[b,i2(p)], reduced over the strip's 16 pair columns,
// and accumulated into per-batch partials for this N-group.
__global__ void __launch_bounds__(32, 1)
honu_gemm(const float* __restrict__ xb,
          const float* __restrict__ Wm,
          const int* __restrict__ pairTab,
          float* __restrict__ partials) {
    const int lane   = threadIdx.x;   // 0..31
    const int lo     = lane & 15;
    const int hi     = lane >> 4;
    const int g      = blockIdx.x;                 // N-group
    const int mbase0 = blockIdx.y * (M_PER_WAVE * 16);
    const int mbase1 = mbase0 + 16;

    // A fragment sources (loop-invariant across N-tiles -> hoisted to VGPRs)
    const float* arow0 = xb + (size_t)(mbase0 + lo) * KPAD + 2 * hi;
    const float* arow1 = xb + (size_t)(mbase1 + lo) * KPAD + 2 * hi;

    float rsum0[8], rsum1[8];
#pragma unroll
    for (int r = 0; r < 8; ++r) { rsum0[r] = 0.0f; rsum1[r] = 0.0f; }

#pragma unroll
    for (int t = 0; t < NT_PER_WAVE; ++t) {
        const int nbase = (g * NT_PER_WAVE + t) * 16;
        const float* brow = Wm + (size_t)(nbase + lo) * KPAD + 2 * hi;

        v8f c0 = {};
        v8f c1 = {};
#pragma unroll
        for (int k0 = 0; k0 < KPAD; k0 += 4) {
            v2f bv = *(const v2f*)(brow + k0);
            v2f a0 = *(const v2f*)(arow0 + k0);
            v2f a1 = *(const v2f*)(arow1 + k0);
            // two independent D = A(16x4) x B(4x16) + C sharing one B fragment
            c0 = __builtin_amdgcn_wmma_f32_16x16x4_f32(
                     false, a0, false, bv, (short)0, c0, false, false);
            c1 = __builtin_amdgcn_wmma_f32_16x16x4_f32(
                     false, a1, false, bv, (short)0, c1, false, false);
        }

        // pair factor: p depends only on the lane's column
        const int e  = pairTab[nbase + lo];
        const int i1 = e & 0xFFFF;
        const int i2 = (e >> 16) & 0xFFFF;

#pragma unroll
        for (int r = 0; r < 8; ++r) {
            const int b0 = mbase0 + r + 8 * hi;   // batch row of c0[r] in this lane
            const int b1 = mbase1 + r + 8 * hi;   // batch row of c1[r] in this lane
            const float f0 = xb[(size_t)b0 * KPAD + i1] * xb[(size_t)b0 * KPAD + i2];
            const float f1 = xb[(size_t)b1 * KPAD + i1] * xb[(size_t)b1 * KPAD + i2];
            float v0 = c0[r] * f0;
            float v1 = c1[r] * f1;
            // reduce over the 16 pair-columns (stays inside each 16-lane half)
            v0 += __shfl_xor(v0, 1);  v1 += __shfl_xor(v1, 1);
            v0 += __shfl_xor(v0, 2);  v1 += __shfl_xor(v1, 2);
            v0 += __shfl_xor(v0, 4);  v1 += __shfl_xor(v1, 4);
            v0 += __shfl_xor(v0, 8);  v1 += __shfl_xor(v1, 8);
            rsum0[r] += v0;
            rsum1[r] += v1;
        }
    }

    if (lo == 0) {                    // lane 0 -> rows +0..7, lane 16 -> rows +8..15
#pragma unroll
        for (int r = 0; r < 8; ++r) {
            partials[(size_t)g * BATCH + mbase0 + r + 8 * hi] = rsum0[r];
            partials[(size_t)g * BATCH + mbase1 + r + 8 * hi] = rsum1[r];
        }
    }
}

// ---------------------------------------------------------------- deterministic final reduce
__global__ void honu_final(const float* __restrict__ partials, float* __restrict__ out) {
    const int b = threadIdx.x;
    float s = 0.0f;
    for (int g = 0; g < NGROUPS; ++g) s += partials[(size_t)g * BATCH + b];
    out[b] = s;
}

// ----------------------------------------------------------------
extern "C" void kernel_launch(void* const* d_in, const int* in_sizes, int n_in,
                              void* d_out, int out_size, void* d_ws, size_t ws_size,
                              hipStream_t stream) {
    const float* x    = (const float*)d_in[0];
    const float* w    = (const float*)d_in[1];
    const int*   cidx = (const int*)d_in[2];
    float*       out  = (float*)d_out;

    float* ws       = (float*)d_ws;
    float* xb       = ws + XB_OFF;
    float* Wm       = ws + W_OFF;
    int*   pairTab  = (int*)(ws + PT_OFF);
    float* partials = ws + PART_OFF;

    const int nM = in_sizes[1];   // number of weights / combinations

    const long nz = WS_FLOATS;
    honu_zero<<<(unsigned)((nz + 255) / 256), 256, 0, stream>>>(ws, nz);
    honu_build_xb<<<BATCH, KPAD, 0, stream>>>(x, xb);
    honu_scatter<<<(nM + 255) / 256, 256, 0, stream>>>(cidx, w, Wm, pairTab, nM);
    honu_gemm<<<dim3(NGROUPS, MGROUPS), 32, 0, stream>>>(xb, Wm, pairTab, partials);
    honu_final<<<1, BATCH, 0, stream>>>(partials, out);
}